// GCNNet_68839735820423
// MI455X (gfx1250) — compile-verified
//
#include <hip/hip_runtime.h>

// GCN 2-layer forward for MI455X (gfx1250, wave32).
// Roofline: only x (205MB) must stream from HBM; t1/acc1/t2/out/edges all fit
// the 192MB L2, so scatter-add atomics and gathers are L2-resident.
// GEMMs use V_WMMA_F32_16X16X4_F32 (exact f32, matches the f32 reference).

#define NODES   100000
#define EDGES   3200000
#define ETOT    (EDGES + NODES)   // edges + self loops
#define FIN     512
#define HID     16
#define CLS     40
#define NTILES  (NODES / 16)      // 6250 (exact)

typedef float v2f __attribute__((ext_vector_type(2)));
typedef float v8f __attribute__((ext_vector_type(8)));

__device__ __forceinline__ void atomAddF(float* p, float v) {
    // hardware global_atomic_add_f32 (coarse-grained memory is fine here)
    unsafeAtomicAdd(p, v);
}

// ---- init: deg=1 (self loop), acc1=0, out=0 -------------------------------
__global__ __launch_bounds__(256) void k_init(float* __restrict__ deg,
                                              float* __restrict__ acc1,
                                              float* __restrict__ out) {
    int i = blockIdx.x * 256 + threadIdx.x;
    if (i < NODES)        deg[i]  = 1.0f;
    if (i < NODES * HID)  acc1[i] = 0.0f;
    if (i < NODES * CLS)  out[i]  = 0.0f;
}

// ---- degree: deg[dst[e]] += 1 ---------------------------------------------
__global__ __launch_bounds__(256) void k_degree(const int* __restrict__ dst,
                                                float* __restrict__ deg) {
    int e = blockIdx.x * 256 + threadIdx.x;
    if (e < EDGES) atomAddF(&deg[dst[e]], 1.0f);
}

// ---- dinv = deg^-1/2 (deg >= 1 always, self loop) -------------------------
__global__ __launch_bounds__(256) void k_dinv(float* __restrict__ deg) {
    int i = blockIdx.x * 256 + threadIdx.x;
    if (i < NODES) deg[i] = 1.0f / sqrtf(deg[i]);
}

// ---- GEMM1: t1[N,16] = x[N,512] @ W1[512,16] via V_WMMA_F32_16X16X4_F32 ---
__global__ __launch_bounds__(256) void k_gemm1(const float* __restrict__ X,
                                               const float* __restrict__ W,
                                               float* __restrict__ T) {
    const int lane = threadIdx.x & 31;
    const int wave = threadIdx.x >> 5;
    const int tile = blockIdx.x * 8 + wave;          // wave-uniform
    if (tile >= NTILES) return;                       // uniform exit: EXEC all-1s below
    const int row0 = tile * 16;
    const int m    = lane & 15;
    const int half = lane >> 4;                       // 0: K=k,k+1  1: K=k+2,k+3
    const int kb   = half * 2;
    const float* xrow = X + (size_t)(row0 + m) * FIN;
    v8f c = {0.f, 0.f, 0.f, 0.f, 0.f, 0.f, 0.f, 0.f};
    for (int k = 0; k < FIN; k += 4) {
        v2f a = *(const v2f*)(xrow + k + kb);                 // A: 16x4 tile
        v2f b;
        b.x = W[(k + kb)     * HID + m];                      // B: 4x16 tile
        b.y = W[(k + kb + 1) * HID + m];
        c = __builtin_amdgcn_wmma_f32_16x16x4_f32(
                false, a, false, b, (short)0, c, false, false);
    }
    float* trow = T + (size_t)row0 * HID;
#pragma unroll
    for (int v = 0; v < 8; ++v) {                             // C/D: VGPR v -> M=v(+8)
        int mr = v + half * 8;
        trow[mr * HID + m] = c[v];
    }
}

// ---- scatter layer1: acc1[d,:] += dinv[s]*dinv[d] * t1[s,:] ---------------
// lane mapping: 16 consecutive lanes = one edge's 16 features (64B coalesced)
__global__ __launch_bounds__(256) void k_scatter1(const int* __restrict__ src,
                                                  const int* __restrict__ dst,
                                                  const float* __restrict__ dinv,
                                                  const float* __restrict__ t1,
                                                  float* __restrict__ acc1) {
    int t = blockIdx.x * 256 + threadIdx.x;
    int e = t >> 4, f = t & 15;
    if (e >= ETOT) return;
    int s, d;
    if (e < EDGES) { s = src[e]; d = dst[e]; } else { s = d = e - EDGES; }
    float nrm = dinv[s] * dinv[d];
    atomAddF(&acc1[d * HID + f], nrm * t1[s * HID + f]);
}

// ---- bias + ReLU (in place on acc1) ---------------------------------------
__global__ __launch_bounds__(256) void k_bias_relu(float* __restrict__ acc1,
                                                   const float* __restrict__ b1) {
    int i = blockIdx.x * 256 + threadIdx.x;
    if (i < NODES * HID) acc1[i] = fmaxf(acc1[i] + b1[i & 15], 0.0f);
}

// ---- GEMM2: t2[N,40] = a1[N,16] @ W2[16,40]; 3 col-tiles (pad 40->48) -----
__global__ __launch_bounds__(256) void k_gemm2(const float* __restrict__ A1,
                                               const float* __restrict__ W2,
                                               float* __restrict__ T2) {
    const int lane  = threadIdx.x & 31;
    const int wave  = threadIdx.x >> 5;
    const int gtile = blockIdx.x * 8 + wave;          // wave-uniform
    if (gtile >= NTILES * 3) return;
    const int rt = gtile / 3, ct = gtile % 3;
    const int row0 = rt * 16, n0 = ct * 16;
    const int m    = lane & 15;
    const int half = lane >> 4;
    const int kb   = half * 2;
    const int col  = n0 + m;
    const bool cok = (col < CLS);
    const float* arow = A1 + (size_t)(row0 + m) * HID;
    v8f c = {0.f, 0.f, 0.f, 0.f, 0.f, 0.f, 0.f, 0.f};
#pragma unroll
    for (int k = 0; k < HID; k += 4) {
        v2f a; a.x = arow[k + kb]; a.y = arow[k + kb + 1];
        v2f b;
        b.x = cok ? W2[(k + kb)     * CLS + col] : 0.0f;
        b.y = cok ? W2[(k + kb + 1) * CLS + col] : 0.0f;
        c = __builtin_amdgcn_wmma_f32_16x16x4_f32(
                false, a, false, b, (short)0, c, false, false);
    }
#pragma unroll
    for (int v = 0; v < 8; ++v) {
        int mr = v + half * 8;
        if (cok) T2[(size_t)(row0 + mr) * CLS + col] = c[v];
    }
}

// ---- scatter layer2: out[d,:] += dinv[s]*dinv[d] * t2[s,:] ----------------
__global__ __launch_bounds__(256) void k_scatter2(const int* __restrict__ src,
                                                  const int* __restrict__ dst,
                                                  const float* __restrict__ dinv,
                                                  const float* __restrict__ t2,
                                                  float* __restrict__ out) {
    int t = blockIdx.x * 256 + threadIdx.x;
    int e = t / CLS, f = t % CLS;
    if (e >= ETOT) return;
    int s, d;
    if (e < EDGES) { s = src[e]; d = dst[e]; } else { s = d = e - EDGES; }
    float nrm = dinv[s] * dinv[d];
    atomAddF(&out[d * CLS + f], nrm * t2[s * CLS + f]);
}

// ---- final bias -----------------------------------------------------------
__global__ __launch_bounds__(256) void k_bias2(float* __restrict__ out,
                                               const float* __restrict__ b2) {
    int i = blockIdx.x * 256 + threadIdx.x;
    if (i < NODES * CLS) out[i] += b2[i % CLS];
}

static inline size_t alignup(size_t x) { return (x + 255) & ~(size_t)255; }

extern "C" void kernel_launch(void* const* d_in, const int* in_sizes, int n_in,
                              void* d_out, int out_size, void* d_ws, size_t ws_size,
                              hipStream_t stream) {
    const float* x  = (const float*)d_in[0];
    const int*   ei = (const int*)d_in[1];
    const float* W1 = (const float*)d_in[2];
    const float* b1 = (const float*)d_in[3];
    const float* W2 = (const float*)d_in[4];
    const float* b2 = (const float*)d_in[5];
    float* out = (float*)d_out;

    const int* srcp = ei;
    const int* dstp = ei + EDGES;

    // workspace carve-up (~27.9 MB)
    char* ws = (char*)d_ws;
    size_t o = 0;
    float* dinv = (float*)(ws + o); o = alignup(o + (size_t)NODES * 4);
    float* t1   = (float*)(ws + o); o = alignup(o + (size_t)NODES * HID * 4);
    float* acc1 = (float*)(ws + o); o = alignup(o + (size_t)NODES * HID * 4);
    float* t2   = (float*)(ws + o); o = alignup(o + (size_t)NODES * CLS * 4);
    (void)ws_size;

    const int B = 256;
    auto blocks = [](long long n) { return (int)((n + 255) / 256); };

    k_init     <<<blocks((long long)NODES * CLS), B, 0, stream>>>(dinv, acc1, out);
    k_degree   <<<blocks(EDGES), B, 0, stream>>>(dstp, dinv);
    k_dinv     <<<blocks(NODES), B, 0, stream>>>(dinv);
    k_gemm1    <<<(NTILES + 7) / 8, B, 0, stream>>>(x, W1, t1);
    k_scatter1 <<<blocks((long long)ETOT * HID), B, 0, stream>>>(srcp, dstp, dinv, t1, acc1);
    k_bias_relu<<<blocks((long long)NODES * HID), B, 0, stream>>>(acc1, b1);
    k_gemm2    <<<(NTILES * 3 + 7) / 8, B, 0, stream>>>(acc1, W2, t2);
    k_scatter2 <<<blocks((long long)ETOT * CLS), B, 0, stream>>>(srcp, dstp, dinv, t2, out);
    k_bias2    <<<blocks((long long)NODES * CLS), B, 0, stream>>>(out, b2);
}